// TemporalConvolution_80307298501386
// MI455X (gfx1250) — compile-verified
//
#include <hip/hip_runtime.h>

// CDNA5 / gfx1250, wave32.
typedef __attribute__((ext_vector_type(2))) float v2f;
typedef __attribute__((ext_vector_type(8))) float v8f;

#define T_DIM 512          // sequence length (fixed by reference)
#define ROWS_PER_WAVE 16
#define WAVES_PER_BLOCK 8
#define ROWS_PER_BLOCK (ROWS_PER_WAVE * WAVES_PER_BLOCK)   // 128

// out[r,f] = sum_t x[r,t] * W[t,f],  W[t,f] = adj[f,t] iff |t-f|==1 else 0.
// Per 16x16 output tile (rows r0..r0+15, cols f0..f0+15) only K-chunks
// c in [4*fi-1, 4*fi+4] (clamped to [0,127]) contribute. Each chunk is one
// V_WMMA_F32_16X16X4_F32.
__global__ __launch_bounds__(256)
void banded_bmm_wmma(const float* __restrict__ x,
                     const float* __restrict__ adj,
                     float* __restrict__ out,
                     int R) {
    __shared__ float s_wl[T_DIM];   // wl[f] = adj[f, f-1]  (0 at f==0)
    __shared__ float s_wh[T_DIM];   // wh[f] = adj[f, f+1]  (0 at f==T-1)
    for (int f = threadIdx.x; f < T_DIM; f += blockDim.x) {
        s_wl[f] = (f > 0)         ? adj[(size_t)f * T_DIM + (f - 1)] : 0.0f;
        s_wh[f] = (f < T_DIM - 1) ? adj[(size_t)f * T_DIM + (f + 1)] : 0.0f;
    }
    __syncthreads();

    const int lane = (int)(threadIdx.x & 31u);
    const int wave = (int)(threadIdx.x >> 5u);
    const int hi   = lane >> 4;      // 0: lanes 0-15, 1: lanes 16-31
    const int l15  = lane & 15;

    const int r0 = ((int)blockIdx.x * WAVES_PER_BLOCK + wave) * ROWS_PER_WAVE;
    if (r0 >= R) return;

    // A fragment: lane holds row M = l15; VGPR j holds K = j + 2*hi.
    const float* xrow = x + (size_t)(r0 + l15) * T_DIM;

    for (int fi = 0; fi < T_DIM / 16; ++fi) {
        const int f0 = fi * 16;
        const int f  = f0 + l15;                 // this lane's output column
        const float wlf = s_wl[f];
        const float whf = s_wh[f];

        v8f acc = {};

        int c_lo = 4 * fi - 1; if (c_lo < 0) c_lo = 0;
        int c_hi = 4 * fi + 4; if (c_hi > T_DIM / 4 - 1) c_hi = T_DIM / 4 - 1;
        for (int c = c_lo; c <= c_hi; ++c) {
            const int t = 4 * c + 2 * hi;        // K for VGPR0 of this lane
            // A: x[row, t], x[row, t+1]  (8B-aligned -> global_load_b64)
            v2f a = *(const v2f*)(xrow + t);
            // B[k, n]: W[t+j, f] synthesized from the two band weights.
            v2f b;
            b.x = (t     == f - 1) ? wlf : ((t     == f + 1) ? whf : 0.0f);
            b.y = (t + 1 == f - 1) ? wlf : ((t + 1 == f + 1) ? whf : 0.0f);
            // 8-arg form: (neg_a, A, neg_b, B, c_mod, C, reuse_a, reuse_b)
            acc = __builtin_amdgcn_wmma_f32_16x16x4_f32(
                false, a, false, b, (short)0, acc, false, false);
        }

        // C/D layout: VGPR v -> row r0 + v + 8*hi, col f.
        const size_t obase = (size_t)(r0 + 8 * hi) * T_DIM + (size_t)f;
#pragma unroll
        for (int v = 0; v < 8; ++v) {
            out[obase + (size_t)v * T_DIM] = acc[v];
        }
    }
}

extern "C" void kernel_launch(void* const* d_in, const int* in_sizes, int n_in,
                              void* d_out, int out_size, void* d_ws, size_t ws_size,
                              hipStream_t stream) {
    (void)n_in; (void)out_size; (void)d_ws; (void)ws_size;
    const float* x   = (const float*)d_in[0];   // [B, N, T] f32
    const float* adj = (const float*)d_in[1];   // [T, T] f32
    float* out = (float*)d_out;                 // [B, N, T] f32

    const int R = in_sizes[0] / T_DIM;          // B*N rows
    const int blocks = (R + ROWS_PER_BLOCK - 1) / ROWS_PER_BLOCK;
    banded_bmm_wmma<<<blocks, 256, 0, stream>>>(x, adj, out, R);
}